// Loss_Function_70368744177806
// MI455X (gfx1250) — compile-verified
//
#include <hip/hip_runtime.h>
#include <hip/hip_bf16.h>
#include <math.h>

// ---------------------------------------------------------------------------
// Circular-GIoU loss, MI455X (gfx1250).
//
// Roofline: ~504 B + ~100 VALU ops per row -> ~23 us of HBM traffic at
// 23.3 TB/s, comparable transcendental VALU time; no matrix structure, so
// WMMA is inapplicable. CDNA5 path: bulk-stage each 256-row slab with
// GLOBAL_LOAD_ASYNC_TO_LDS_B128 (ASYNCcnt) using literal offsets (the
// 24-bit INST_OFFSET is added to BOTH the LDS and global addresses, and this
// is a contiguous copy, so one base address pair serves all rounds), drain
// with s_wait_asynccnt, then compute from conflict-free LDS and write
// non-temporal vectorized stores.
// ---------------------------------------------------------------------------

#define RPB       256   // rows per block (= threads per block, 8 waves)
#define PRED_COLS 26
#define TGT_COLS  50
#define NJ        24

typedef __attribute__((ext_vector_type(4))) float v4f;   // native clang vector

static constexpr float PI_F    = 3.14159265358979323846f;
static constexpr float EPS_DIV = 1e-8f;
static constexpr float EPS_IOU = 1e-6f;

// One async 16B-per-lane copy round; OFF must be a literal (applied to both
// the LDS destination and the global source address).
#define ASYNC_B128(la, ga, OFF)                                              \
    asm volatile("global_load_async_to_lds_b128 %0, %1, off offset:" #OFF    \
                 :: "v"(la), "v"(ga) : "memory")

__global__ __launch_bounds__(RPB) void giou_circle_kernel(
    const float* __restrict__ pred,    // [N, 26]
    const float* __restrict__ target,  // [N, 50]
    float* __restrict__ out,           // loss[N*24] | pd_cx[N] | pd_cy[N] | scale_pd[N*24]
    int N)
{
    __shared__ __align__(16) float sP[RPB * PRED_COLS];  // 26 KiB
    __shared__ __align__(16) float sT[RPB * TGT_COLS];   // 50 KiB

    const int       tid     = threadIdx.x;
    const long long rowBase = (long long)blockIdx.x * RPB;

    // 32-bit wave-relative LDS byte offsets (low 32 bits of the generic ptr).
    const unsigned sPoff = (unsigned)(size_t)(void*)&sP[0];
    const unsigned sToff = (unsigned)(size_t)(void*)&sT[0];

    const bool full = (rowBase + RPB) <= (long long)N;   // uniform branch

    if (full) {
        // ---- fast path: unguarded B128 staging, immediate offsets --------
        // pred slab  = 26624 B = 6*4096 + 128 lanes * 16 B
        // target slab= 51200 B = 12*4096 + 128 lanes * 16 B
        const unsigned           laP = sPoff + (unsigned)tid * 16u;
        const unsigned           laT = sToff + (unsigned)tid * 16u;
        const unsigned long long gaP =
            (unsigned long long)(size_t)(pred + rowBase * PRED_COLS) + (unsigned)tid * 16u;
        const unsigned long long gaT =
            (unsigned long long)(size_t)(target + rowBase * TGT_COLS) + (unsigned)tid * 16u;

        ASYNC_B128(laP, gaP, 0);
        ASYNC_B128(laP, gaP, 4096);
        ASYNC_B128(laP, gaP, 8192);
        ASYNC_B128(laP, gaP, 12288);
        ASYNC_B128(laP, gaP, 16384);
        ASYNC_B128(laP, gaP, 20480);

        ASYNC_B128(laT, gaT, 0);
        ASYNC_B128(laT, gaT, 4096);
        ASYNC_B128(laT, gaT, 8192);
        ASYNC_B128(laT, gaT, 12288);
        ASYNC_B128(laT, gaT, 16384);
        ASYNC_B128(laT, gaT, 20480);
        ASYNC_B128(laT, gaT, 24576);
        ASYNC_B128(laT, gaT, 28672);
        ASYNC_B128(laT, gaT, 32768);
        ASYNC_B128(laT, gaT, 36864);
        ASYNC_B128(laT, gaT, 40960);
        ASYNC_B128(laT, gaT, 45056);

        if (tid < 128) {                 // half rounds (2048 B each)
            ASYNC_B128(laP, gaP, 24576);
            ASYNC_B128(laT, gaT, 49152);
        }
    } else {
        // ---- tail path (N % 256 != 0): guarded per-element B32 copies ----
        {
            const long long base = rowBase * PRED_COLS;
            const long long tot  = (long long)N * PRED_COLS;
            for (int k = 0; k < PRED_COLS; ++k) {
                long long idx = base + (long long)k * RPB + tid;
                if (idx < tot) {
                    unsigned long long ga = (unsigned long long)(size_t)(pred + idx);
                    unsigned           la = sPoff + (unsigned)((k * RPB + tid) * 4);
                    asm volatile("global_load_async_to_lds_b32 %0, %1, off"
                                 :: "v"(la), "v"(ga) : "memory");
                }
            }
        }
        {
            const long long base = rowBase * TGT_COLS;
            const long long tot  = (long long)N * TGT_COLS;
            for (int k = 0; k < TGT_COLS; ++k) {
                long long idx = base + (long long)k * RPB + tid;
                if (idx < tot) {
                    unsigned long long ga = (unsigned long long)(size_t)(target + idx);
                    unsigned           la = sToff + (unsigned)((k * RPB + tid) * 4);
                    asm volatile("global_load_async_to_lds_b32 %0, %1, off"
                                 :: "v"(la), "v"(ga) : "memory");
                }
            }
        }
    }

    // Drain this wave's async copies, then make them visible block-wide.
    asm volatile("s_wait_asynccnt 0" ::: "memory");
    __syncthreads();

    // ---- per-row compute (stride-26/50 word reads across 32 lanes are
    //      injective mod 64 banks -> conflict-free) ------------------------
    const long long r = rowBase + tid;
    if (r < (long long)N) {
        const float* P = &sP[tid * PRED_COLS];
        const float* T = &sT[tid * TGT_COLS];

        const float pcx = P[0], pcy = P[1];
        const float gcx = T[0], gcy = T[1];
        const float ddx = gcx - pcx, ddy = gcy - pcy;
        const float d  = sqrtf(ddx * ddx + ddy * ddy);
        const float d2 = d * d;

        float loss[NJ];
        float spd[NJ];

        #pragma unroll
        for (int j = 0; j < NJ; ++j) {
            const float vx = T[2 + 2 * j] - gcx;
            const float vy = T[3 + 2 * j] - gcy;
            const float sg = sqrtf(vx * vx + vy * vy);
            const float sp = P[2 + j];
            spd[j] = sp;

            const float ag  = PI_F * sg * sg;
            const float ap  = PI_F * sp * sp;
            const float mn  = fminf(sg, sp);
            const float mx  = fmaxf(sg, sp);
            const float mn2 = mn * mn;
            const float mx2 = mx * mx;

            float acmin = (mn2 + d2 - mx2) / (2.0f * mn * d + EPS_DIV);
            float acmax = (mx2 + d2 - mn2) / (2.0f * mx * d + EPS_DIV);
            acmin = fminf(fmaxf(acmin, -0.99f), 0.99f);
            acmax = fminf(fmaxf(acmax, -0.99f), 0.99f);

            const float am = acosf(acmin);
            const float aM = acosf(acmax);

            const float inter = am * mn2 + aM * mx2 - mn * d * sinf(am);

            const bool contained = fabsf(sg - sp) >= d;
            const bool disjoint  = d >= (sg + sp);

            const float resi = disjoint ? 0.0f
                               : (contained ? PI_F * mn2 : inter);
            const float uni  = ag + ap - resi;
            const float iou  = resi / (uni + EPS_IOU);

            const float cl  = contained ? mx : 0.5f * (sg + sp + d);
            const float cs  = PI_F * cl * cl;
            const float top = cs - uni;

            loss[j] = 1.0f - (iou - top / cs);
        }

        // ---- non-temporal stores: pure streaming output, skip L2 rinse ----
        float* lo = out + r * (long long)NJ;          // 96-B row stride, 16-B aligned
        #pragma unroll
        for (int i = 0; i < NJ / 4; ++i) {
            v4f v = { loss[4*i], loss[4*i+1], loss[4*i+2], loss[4*i+3] };
            __builtin_nontemporal_store(v, (v4f*)lo + i);
        }

        __builtin_nontemporal_store(pcx, out + (long long)N * NJ       + r);
        __builtin_nontemporal_store(pcy, out + (long long)N * (NJ + 1) + r);

        float* so = out + (long long)N * (NJ + 2) + r * (long long)NJ;
        if ((((size_t)so) & 15) == 0) {               // uniform; true for even N
            #pragma unroll
            for (int i = 0; i < NJ / 4; ++i) {
                v4f v = { spd[4*i], spd[4*i+1], spd[4*i+2], spd[4*i+3] };
                __builtin_nontemporal_store(v, (v4f*)so + i);
            }
        } else {
            #pragma unroll
            for (int i = 0; i < NJ; ++i)
                __builtin_nontemporal_store(spd[i], so + i);
        }
    }
}

extern "C" void kernel_launch(void* const* d_in, const int* in_sizes, int n_in,
                              void* d_out, int out_size, void* d_ws, size_t ws_size,
                              hipStream_t stream)
{
    const float* pred   = (const float*)d_in[0];
    const float* target = (const float*)d_in[1];
    float*       out    = (float*)d_out;

    const int N      = in_sizes[0] / PRED_COLS;   // 1048576
    const int blocks = (N + RPB - 1) / RPB;       // 4096

    giou_circle_kernel<<<blocks, RPB, 0, stream>>>(pred, target, out, N);
}